// TriplanarDecoder_8100308320685
// MI455X (gfx1250) — compile-verified
//
#include <hip/hip_runtime.h>
#include <hip/hip_bf16.h>

typedef __attribute__((ext_vector_type(2))) float v2f;
typedef __attribute__((ext_vector_type(8))) float v8f;

#define NPTS            524288   // 4 * 1024 * 128 query points
#define TILES           32768    // NPTS / 16
#define TPW             4        // tiles per wave
#define WAVES_PER_BLOCK 8
#define BLOCK           256

// Unaligned-safe 8-byte load (pair address may be only 4-byte aligned).
__device__ __forceinline__ v2f load2_u(const float* p) {
  v2f r;
  __builtin_memcpy(&r, p, sizeof(r));
  return r;
}

// Bilinear grid-sample (border clamp) of this lane's 16 channels from one
// 256x256x32 plane; accumulates into val[]. Channel mapping puts results
// directly into WMMA A-fragment order: val[2s+j] = x[m][4s + 2*half + j].
//
// Border handling is folded into the weights: the horizontal pair is loaded
// at ix0L = min(ix0, 254) with wx' = ix - ix0L, so the clamped case
// (ix == 255.0) gives wx' == 1.0 and the lerp lands on the border texel.
// Same trick vertically with iy0L. No selects needed.
__device__ __forceinline__ void sample_plane_add(const float* __restrict__ base,
                                                 float gx, float gy, int half,
                                                 float* val) {
  float ix = fminf(fmaxf((gx + 1.0f) * 0.5f * 255.0f, 0.0f), 255.0f);
  float iy = fminf(fmaxf((gy + 1.0f) * 0.5f * 255.0f, 0.0f), 255.0f);
  int ix0L = min((int)floorf(ix), 254);
  int iy0L = min((int)floorf(iy), 254);
  float wx = ix - (float)ix0L;   // in [0,1]; ==1.0 exactly at right border
  float wy = iy - (float)iy0L;   // in [0,1]; ==1.0 exactly at bottom border
  int o0 = (iy0L << 8) + ix0L;   // top row pair offset
  int o1 = o0 + 256;             // bottom row pair offset
#pragma unroll
  for (int i = 0; i < 16; ++i) {
    int c = ((i >> 1) << 2) + (half << 1) + (i & 1);
    const float* cp = base + ((size_t)c << 16);  // channel stride 256*256
    v2f p0 = load2_u(cp + o0);   // {v00, v01}
    v2f p1 = load2_u(cp + o1);   // {v10, v11}
    float top = p0.x + wx * (p0.y - p0.x);
    float bot = p1.x + wx * (p1.y - p1.x);
    val[i] += top + wy * (bot - top);
  }
}

__global__ __launch_bounds__(BLOCK) void triplanar_wmma_kernel(
    const float* __restrict__ xy, const float* __restrict__ xz,
    const float* __restrict__ yz, const float* __restrict__ coords,
    const float* __restrict__ w1, const float* __restrict__ b1,
    const float* __restrict__ w2, const float* __restrict__ b2,
    float* __restrict__ out) {
  __shared__ __attribute__((aligned(16))) float w1s[64 * 32];
  __shared__ __attribute__((aligned(16))) float w2s[4 * 64];
  __shared__ __attribute__((aligned(16))) float b1s[64];
  __shared__ __attribute__((aligned(16))) float b2s[4];
  __shared__ __attribute__((aligned(16))) float hs_all[WAVES_PER_BLOCK * 16 * 64];

  // Stage weights into LDS once per block.
  for (int i = threadIdx.x; i < 64 * 32; i += BLOCK) w1s[i] = w1[i];
  for (int i = threadIdx.x; i < 4 * 64; i += BLOCK) w2s[i] = w2[i];
  if (threadIdx.x < 64) b1s[threadIdx.x] = b1[threadIdx.x];
  if (threadIdx.x < 4)  b2s[threadIdx.x] = b2[threadIdx.x];
  __syncthreads();

  const int lane = threadIdx.x & 31;
  const int wave = threadIdx.x >> 5;
  const int m = lane & 15;       // point-in-tile (A rows) / N index (B cols)
  const int half = lane >> 4;    // K-half selector of f32 WMMA fragments
  float* hs = hs_all + wave * (16 * 64);

  const int gw = blockIdx.x * WAVES_PER_BLOCK + wave;

  const float S1 = 0.17677669529663689f;  // 1/sqrt(32)
  const float S2 = 0.125f;                // 1/sqrt(64)
  const v8f vzero = {};

  for (int it = 0; it < TPW; ++it) {
    const int tile = gw * TPW + it;
    const int p0 = tile << 4;
    const int p = p0 + m;          // this lane's point (dup across halves)
    const int b = p >> 17;         // batch = p / (1024*128)

    const float c0 = coords[3 * p + 0];
    const float c1 = coords[3 * p + 1];
    const float c2 = coords[3 * p + 2];
    const size_t pb = (size_t)b * (32u * 65536u);

    float val[16];
#pragma unroll
    for (int i = 0; i < 16; ++i) val[i] = 0.0f;
    sample_plane_add(xy + pb, c0, c1, half, val);
    sample_plane_add(xz + pb, c0, c2, half, val);
    sample_plane_add(yz + pb, c1, c2, half, val);
#pragma unroll
    for (int i = 0; i < 16; ++i) val[i] *= (1.0f / 3.0f);

    // ---- Layer 1: h(16x64) = x(16x32) * w1^T, via f32 WMMA, K in steps of 4.
    v8f hacc[4];
#pragma unroll
    for (int t = 0; t < 4; ++t) hacc[t] = vzero;
#pragma unroll
    for (int t = 0; t < 4; ++t) {
      const float* wrow = w1s + (16 * t + m) * 32 + 2 * half;  // B: K=4s+2h+j, N=16t+m
#pragma unroll
      for (int s = 0; s < 8; ++s) {
        v2f a;
        a.x = val[2 * s];
        a.y = val[2 * s + 1];
        v2f bb = *(const v2f*)(wrow + 4 * s);  // ds_load_b64 (8B aligned)
        hacc[t] = __builtin_amdgcn_wmma_f32_16x16x4_f32(
            false, a, false, bb, (short)0, hacc[t], false, false);
      }
    }

    // ---- softplus(s1*h + b1) and transpose h into LDS as [point][hidden].
#pragma unroll
    for (int t = 0; t < 4; ++t) {
      float bias = b1s[16 * t + m];
#pragma unroll
      for (int v = 0; v < 8; ++v) {
        float x = hacc[t][v] * S1 + bias;
        float sp = fmaxf(x, 0.0f) + log1pf(__expf(-fabsf(x)));
        hs[(v + 8 * half) * 64 + 16 * t + m] = sp;  // C/D layout -> [pt][hid]
      }
    }
    // Intra-wave LDS RAW fence: lanes read values other lanes wrote.
    asm volatile("s_wait_dscnt 0x0" ::: "memory");

    // ---- Layer 2: out(16x16) = h(16x64) * w2^T (N padded 4->16), K=64.
    v8f oacc = vzero;
    const float* hrow = hs + m * 64 + 2 * half;            // A: h[m][4s+2h+j]
    const float* w2row = w2s + (m & 3) * 64 + 2 * half;    // B: w2[n][4s+2h+j]
#pragma unroll
    for (int s = 0; s < 16; ++s) {
      v2f a2 = *(const v2f*)(hrow + 4 * s);
      v2f bb = *(const v2f*)(w2row + 4 * s);
      oacc = __builtin_amdgcn_wmma_f32_16x16x4_f32(
          false, a2, false, bb, (short)0, oacc, false, false);
    }

    // ---- Epilogue: col 0 -> density, cols 1..3 -> features.
    if (m < 4) {
      float bias = b2s[m];
#pragma unroll
      for (int v = 0; v < 8; ++v) {
        float o = oacc[v] * S2 + bias;
        int pt = p0 + v + 8 * half;
        if (m == 0)
          out[3 * NPTS + pt] = o;          // density_or_distance
        else
          out[3 * pt + (m - 1)] = o;       // features
      }
    }
  }
}

extern "C" void kernel_launch(void* const* d_in, const int* in_sizes, int n_in,
                              void* d_out, int out_size, void* d_ws, size_t ws_size,
                              hipStream_t stream) {
  const float* xy     = (const float*)d_in[0];
  const float* xz     = (const float*)d_in[1];
  const float* yz     = (const float*)d_in[2];
  const float* coords = (const float*)d_in[3];
  const float* w1     = (const float*)d_in[4];
  const float* b1     = (const float*)d_in[5];
  const float* w2     = (const float*)d_in[6];
  const float* b2     = (const float*)d_in[7];
  float* out = (float*)d_out;

  const int blocks = TILES / (WAVES_PER_BLOCK * TPW);  // 1024
  triplanar_wmma_kernel<<<blocks, BLOCK, 0, stream>>>(
      xy, xz, yz, coords, w1, b1, w2, b2, out);
}